// AdaptiveFourierFeatures_44693429682708
// MI455X (gfx1250) — compile-verified
//
#include <hip/hip_runtime.h>
#include <math.h>

#define BB 8
#define SS 2048
#define DD 64
#define FFC 16
#define AAC 32
#define HHC 4
#define OOC 64
#define DIN 2112            // D + 2*D*F
#define NTOK (BB*SS)        // 16384
#define NCOL 128            // 64 gate cols + 64 proj cols
#define CI_STRIDE 2114      // 2112 padded (+2) -> LDS bank-conflict-free, 8B aligned rows
#define INV_SQRT_HD 0.35355339059327373f

typedef float v2f __attribute__((ext_vector_type(2)));
typedef float v8f __attribute__((ext_vector_type(8)));

// ---- workspace layout (float offsets) ----
#define WS_SC    0u                                     // S*2048 sincos table (16 MB, L2-resident)
#define WS_WC    (WS_SC + (size_t)SS*2048)              // + 270336 combined [Wg;Wp]
#define WS_WAL   (WS_WC + (size_t)NCOL*DIN)             // + 256 Walpha[4][64]
#define WS_CAL   (WS_WAL + 256)                         // + 4   calpha[4]

// Fold the whole attention-score pipeline into Walpha[4][64], calpha[4].
// (keys are affine in f_df; the f-constant part cancels in softmax, so the
//  entire MHA reduces to alpha[h] = Walpha[h].x + calpha[h] per token.)
__global__ void prep_kernel(const float* __restrict__ Wq, const float* __restrict__ bq,
                            const float* __restrict__ Wk1,
                            const float* __restrict__ Wqi, const float* __restrict__ bqi,
                            const float* __restrict__ Wki,
                            float* __restrict__ wal, float* __restrict__ cal) {
  __shared__ float u[32];
  __shared__ float w2[128];
  int tid = threadIdx.x;
  if (tid < 32) {                      // u = Wki @ Wk1
    float acc = 0.f;
    for (int a = 0; a < 32; ++a) acc += Wki[tid*32 + a] * Wk1[a];
    u[tid] = acc;
  }
  __syncthreads();
  if (tid < 128) {                     // w2[h][a] = sum_e u[8h+e] * Wqi[8h+e][a]
    int h = tid >> 5, a = tid & 31;
    float acc = 0.f;
    for (int e = 0; e < 8; ++e) acc += u[h*8 + e] * Wqi[(h*8 + e)*32 + a];
    w2[tid] = acc;
  }
  __syncthreads();
  {                                    // Walpha[h][d] = (1/sqrt8) sum_a w2[h][a] * Wq[a][d]
    int h = tid >> 6, d = tid & 63;
    float acc = 0.f;
    for (int a = 0; a < 32; ++a) acc += w2[h*32 + a] * Wq[a*64 + d];
    wal[tid] = acc * INV_SQRT_HD;
  }
  if (tid < 4) {
    float acc = 0.f;
    for (int a = 0; a < 32; ++a) acc += w2[tid*32 + a] * bq[a];
    for (int e = 0; e < 8; ++e)  acc += u[tid*8 + e] * bqi[tid*8 + e];
    cal[tid] = acc * INV_SQRT_HD;
  }
}

// Combined GEMM B matrix: row n = (n<64 ? Wg[n] : Wp[n-64]), contiguous over K.
__global__ void wcomb_kernel(const float* __restrict__ Wg, const float* __restrict__ Wp,
                             float* __restrict__ wc) {
  int e = blockIdx.x * 256 + threadIdx.x;
  if (e >= NCOL * DIN) return;
  int n = e / DIN, i = e - n * DIN;
  wc[e] = (n < OOC) ? Wg[n*DIN + i] : Wp[(n - OOC)*DIN + i];
}

// sc[s][d*32 + j] : j<16 -> sin(2pi t f + ph), j>=16 -> cos(...). Reused across batch.
__global__ void sincos_kernel(const float* __restrict__ fm, const float* __restrict__ fs,
                              const float* __restrict__ ph, float* __restrict__ sc) {
  int e = blockIdx.x * 256 + threadIdx.x;      // S*2048 = 4194304
  int s = e >> 11, i = e & 2047;
  int d = i >> 5, j = i & 31, f = j & 15;
  float fr  = fm[d*16 + f] * fs[d*16 + f];
  float ang = 6.283185307179586f * ((float)s * (1.0f/2047.0f)) * fr + ph[d*16 + f];
  sc[e] = (j & 16) ? cosf(ang) : sinf(ang);
}

// Fused: per 16-token tile -> alpha, softmax attn weights, weighted fourier
// features (ci) in LDS, f32 WMMA GEMM vs [Wg;Wp], sigmoid*silu + residual.
__global__ void __launch_bounds__(256) main_kernel(
    const float* __restrict__ x, const float* __restrict__ fmat,
    const float* __restrict__ fscale,
    const float* __restrict__ bg, const float* __restrict__ bpr,
    const float* __restrict__ wal, const float* __restrict__ cal,
    const float* __restrict__ sc, const float* __restrict__ wc,
    float* __restrict__ out) {
  extern __shared__ float smem[];
  float* ci   = smem;                        // 16 * CI_STRIDE
  float* fmS  = smem + 16 * CI_STRIDE;       // 1024
  float* alS  = fmS + 1024;                  // 64
  float* walS = alS + 64;                    // 256
  float* calS = walS + 256;                  // 4
  const int tid  = threadIdx.x;
  const int base = blockIdx.x * 16;          // token base

  // ---- phase 0: stage fm, Walpha, x-part of ci ----
  #pragma unroll
  for (int r = 0; r < 4; ++r) {
    int i = tid + r*256;                     // 1024
    fmS[i] = fmat[i] * fscale[i];
  }
  walS[tid] = wal[tid];
  if (tid < 4) calS[tid] = cal[tid];
  #pragma unroll
  for (int r = 0; r < 4; ++r) {
    int i = tid + r*256;                     // 16 tok * 64
    int t = i >> 6, d = i & 63;
    ci[t*CI_STRIDE + d] = x[(size_t)(base + t)*DD + d];
  }
  __syncthreads();

  // ---- phase 0.5: alpha[t][h] = Walpha[h] . x[t] + calpha[h] (from LDS) ----
  if (tid < 64) {
    int t = tid >> 2, h = tid & 3;
    float acc = calS[h];
    #pragma unroll
    for (int d = 0; d < DD; ++d) acc += walS[h*64 + d] * ci[t*CI_STRIDE + d];
    alS[tid] = acc;                          // tid == t*4 + h
  }
  __syncthreads();

  // ---- phase 1: head-avg softmax attn weights + weighted fourier features ----
  #pragma unroll
  for (int r = 0; r < 4; ++r) {
    int p = tid + r*256;                     // 1024 (t,d) pairs
    int t = p >> 6, d = p & 63;
    const float* fmd = &fmS[d*16];
    float aw[16];
    #pragma unroll
    for (int f = 0; f < 16; ++f) aw[f] = 0.f;
    #pragma unroll
    for (int h = 0; h < 4; ++h) {
      float a = alS[t*4 + h];
      float sv[16];
      float m = -3.4e38f;
      #pragma unroll
      for (int f = 0; f < 16; ++f) { sv[f] = a * fmd[f]; m = fmaxf(m, sv[f]); }
      float sum = 0.f;
      #pragma unroll
      for (int f = 0; f < 16; ++f) { sv[f] = __expf(sv[f] - m); sum += sv[f]; }
      float inv = 0.25f / sum;               // includes 1/H
      #pragma unroll
      for (int f = 0; f < 16; ++f) aw[f] += sv[f] * inv;
    }
    int s = (base + t) & (SS - 1);           // token % S (S power of two)
    const float* scp = sc + (size_t)s*2048 + d*32;
    float* cip = ci + t*CI_STRIDE + 64 + d*32;
    #pragma unroll
    for (int f = 0; f < 16; ++f) {
      cip[f]      = scp[f]      * aw[f];     // sin part
      cip[16 + f] = scp[16 + f] * aw[f];     // cos part
    }
  }
  __syncthreads();

  // ---- phase 2: f32 WMMA GEMM: [16 x 2112] x [2112 x 128] ----
  // A frag: lane L holds M=L%16, K=2*(L/16)+{0,1}; B frag mirrors with N=L%16.
  const int wave  = tid >> 5;
  const int lane  = tid & 31;
  const int n16   = lane & 15;
  const int khalf = (lane >> 4) << 1;        // {0,2}
  const float* ap  = ci + n16*CI_STRIDE + khalf;                 // A: token row
  const float* bpp = wc + (size_t)(wave*16 + n16)*DIN + khalf;   // B: column row
  v8f acc = {};
  for (int k = 0; k < DIN; k += 16) {
    #pragma unroll
    for (int kk = 0; kk < 16; kk += 4) {
      v2f av = *(const v2f*)(ap + k + kk);
      v2f bv = *(const v2f*)(bpp + k + kk);
      acc = __builtin_amdgcn_wmma_f32_16x16x4_f32(
          /*neg_a=*/false, av, /*neg_b=*/false, bv,
          /*c_mod=*/(short)0, acc, /*reuse_a=*/false, /*reuse_b=*/false);
    }
  }
  __syncthreads();                           // all ci reads done before reuse as res

  // ---- phase 3: spill 16x128 result tile to LDS ----
  float* res = smem;                         // 2048 floats, aliases ci
  #pragma unroll
  for (int r = 0; r < 8; ++r) {
    int row = r + ((lane >> 4) << 3);        // C layout: M = r + 8*(lane/16)
    res[row*NCOL + wave*16 + n16] = acc[r];
  }
  __syncthreads();

  // ---- phase 4: bias + sigmoid*silu + residual ----
  #pragma unroll
  for (int r = 0; r < 4; ++r) {
    int q = tid + r*256;                     // 16*64
    int t = q >> 6, n = q & 63;
    float g  = res[t*NCOL + n]      + bg[n];
    float pv = res[t*NCOL + 64 + n] + bpr[n];
    float sig = 1.f / (1.f + __expf(-g));
    float si  = pv / (1.f + __expf(-pv));
    size_t o = (size_t)(base + t)*DD + n;
    out[o] = x[o] + sig * si;
  }
}

extern "C" void kernel_launch(void* const* d_in, const int* in_sizes, int n_in,
                              void* d_out, int out_size, void* d_ws, size_t ws_size,
                              hipStream_t stream) {
  const float* x    = (const float*)d_in[0];
  const float* fmat = (const float*)d_in[1];
  const float* ph   = (const float*)d_in[2];
  const float* fsc  = (const float*)d_in[3];
  const float* Wq   = (const float*)d_in[4];
  const float* bq   = (const float*)d_in[5];
  const float* Wk1  = (const float*)d_in[6];
  // d_in[7] = bk1, d_in[11] = bki: provably cancel in the softmax -> unused
  const float* Wqi  = (const float*)d_in[8];
  const float* bqi  = (const float*)d_in[9];
  const float* Wki  = (const float*)d_in[10];
  const float* Wg   = (const float*)d_in[12];
  const float* bg   = (const float*)d_in[13];
  const float* Wp   = (const float*)d_in[14];
  const float* bp   = (const float*)d_in[15];
  float* out = (float*)d_out;

  float* ws  = (float*)d_ws;
  float* sc  = ws + WS_SC;
  float* wc  = ws + WS_WC;
  float* wal = ws + WS_WAL;
  float* cal = ws + WS_CAL;

  prep_kernel<<<1, 256, 0, stream>>>(Wq, bq, Wk1, Wqi, bqi, Wki, wal, cal);
  wcomb_kernel<<<(NCOL*DIN + 255)/256, 256, 0, stream>>>(Wg, Wp, wc);
  sincos_kernel<<<(SS*2048)/256, 256, 0, stream>>>(fmat, fsc, ph, sc);

  // LDS: ci(16*2114) + fm(1024) + alpha(64) + Walpha(256) + calpha(4)
  size_t smem_bytes = (size_t)(16*CI_STRIDE + 1024 + 64 + 256 + 4) * sizeof(float);
  main_kernel<<<NTOK/16, 256, smem_bytes, stream>>>(x, fmat, fsc, bg, bp, wal, cal, sc, wc, out);
}